// LocalModel_Layer_35493609734520
// MI455X (gfx1250) — compile-verified
//
#include <hip/hip_runtime.h>
#include <stdint.h>

// out[n] = dot(x[n,:], W[idx[n],:])  -- fused gather-dot, memory-bound.
// N = 262144, D = 256, C = 256, all f32 (idx delivered as int32 by harness).

typedef float v2f __attribute__((ext_vector_type(2)));
typedef float v8f __attribute__((ext_vector_type(8)));

#define N_ROWS        262144
#define DDIM          256
#define CDIM          256
#define LDS_STRIDE    258   // padded W row stride in floats: 1032 B, 8B-aligned, bank step 2
#define WG_THREADS    512
#define ROWS_PER_WAVE 16
#define ROWS_PER_WG   (16 * ROWS_PER_WAVE)   // 16 waves * 16 rows = 256

__global__ void __launch_bounds__(WG_THREADS, 1)
gather_dot_wmma(const float* __restrict__ x,
                const int*   __restrict__ idx,
                const float* __restrict__ W,
                float*       __restrict__ out)
{
    extern __shared__ float wlds[];   // CDIM * LDS_STRIDE floats (~258 KB)

    const int tid = threadIdx.x;

    // ---------------- Stage all of W into padded LDS via gfx1250 async copies ----------------
    {
        const uint32_t ldsBase = (uint32_t)(uintptr_t)wlds;
        const uint64_t wbase   = (uint64_t)W;
        // copy in float2 (b64) units: 256 rows * 128 f2 = 32768 units, 64 per thread
        #pragma unroll 4
        for (int i = tid; i < CDIM * (DDIM / 2); i += WG_THREADS) {
            const int row = i >> 7;          // / (DDIM/2)
            const int c2  = i & 127;
            const uint32_t voff  = (uint32_t)(row * (DDIM * 4) + c2 * 8);        // global byte off
            const uint32_t laddr = ldsBase + (uint32_t)(row * (LDS_STRIDE * 4) + c2 * 8);
            asm volatile("global_load_async_to_lds_b64 %0, %1, %2"
                         :: "v"(laddr), "v"(voff), "s"(wbase)
                         : "memory");
        }
        asm volatile("s_wait_asynccnt 0" ::: "memory");
    }
    __syncthreads();

    // ---------------- WMMA gathered row-dot: 16 rows per wave ----------------
    const int wave = tid >> 5;
    const int lane = tid & 31;
    const int r16  = lane & 15;      // row (for A) / column (for B) within the 16x16 tile
    const int half = lane >> 4;      // K sub-offset select per f32 WMMA operand layout

    const int n0 = blockIdx.x * ROWS_PER_WG + wave * ROWS_PER_WAVE;

    // A: lane supplies x[n0+r16][4kc+2h .. +1]
    const float* xrow = x + (size_t)(n0 + r16) * DDIM + 2 * half;
    // B: lane supplies W[idx[n0+r16]][4kc+2h .. +1] from LDS
    const int cls = idx[n0 + r16];
    const float* brow = wlds + cls * LDS_STRIDE + 2 * half;

    v8f acc = {};
    #pragma unroll 8
    for (int kc = 0; kc < DDIM / 4; ++kc) {
        v2f a = *(const v2f*)(xrow + 4 * kc);
        v2f b = *(const v2f*)(brow + 4 * kc);
        // (neg_a, A, neg_b, B, c_mod, C, reuse_a, reuse_b)
        acc = __builtin_amdgcn_wmma_f32_16x16x4_f32(false, a, false, b,
                                                    (short)0, acc, false, false);
    }

    // ---------------- Extract diagonal D[r][r] ----------------
    // C/D layout: VGPR v, lanes 0-15 -> (M=v, N=lane); lanes 16-31 -> (M=v+8, N=lane-16).
    // Diagonal: lanes 0..7 hold r=lane in acc[lane]; lanes 24..31 hold r=lane-16 in acc[lane-24].
    int vsel = -1, row = -1;
    if (lane < 8)        { vsel = lane;      row = lane;      }
    else if (lane >= 24) { vsel = lane - 24; row = lane - 16; }
    if (vsel >= 0) {
        float d = acc[0];
        #pragma unroll
        for (int j = 1; j < 8; ++j) {
            if (vsel == j) d = acc[j];
        }
        out[n0 + row] = d;
    }
}

extern "C" void kernel_launch(void* const* d_in, const int* in_sizes, int n_in,
                              void* d_out, int out_size, void* d_ws, size_t ws_size,
                              hipStream_t stream)
{
    const float* x   = (const float*)d_in[0];
    const int*   idx = (const int*)  d_in[1];
    const float* W   = (const float*)d_in[2];
    float*       out = (float*)d_out;

    const size_t shmem = (size_t)CDIM * LDS_STRIDE * sizeof(float);  // 264192 B < 320 KB
    (void)hipFuncSetAttribute((const void*)gather_dot_wmma,
                              hipFuncAttributeMaxDynamicSharedMemorySize, (int)shmem);

    dim3 grid(N_ROWS / ROWS_PER_WG);   // 1024
    dim3 block(WG_THREADS);            // 512 = 16 wave32
    gather_dot_wmma<<<grid, block, shmem, stream>>>(x, idx, W, out);
}